// GCNLayer_1151051235745
// MI455X (gfx1250) — compile-verified
//
#include <hip/hip_runtime.h>

typedef __attribute__((ext_vector_type(2))) float v2f;
typedef __attribute__((ext_vector_type(8))) float v8f;

#define N_NODES 50000
#define N_EDGES 600000
#define D 128
#define MT_COUNT (N_NODES / 16)   // 3125 row tiles, exact

// ---------------------------------------------------------------------------
// Kernel 1: h = feature  (residual term folded into the scatter accumulator)
// ---------------------------------------------------------------------------
__global__ void gcn_init_h(const float4* __restrict__ feat4,
                           float4* __restrict__ h4, int n4) {
    int i = blockIdx.x * blockDim.x + threadIdx.x;
    if (i < n4) h4[i] = feat4[i];
}

// ---------------------------------------------------------------------------
// Kernel 2: h[dst[e]] += feature[src[e]]  — one wave32 per edge, 4 f32/lane,
// native global_atomic_add_f32 (L2-resident: feature+h are 51 MB on 192 MB L2)
// ---------------------------------------------------------------------------
__global__ void gcn_scatter(const float* __restrict__ feature,
                            const int* __restrict__ src,
                            const int* __restrict__ dst,
                            float* __restrict__ h) {
    int e    = blockIdx.x * (blockDim.x >> 5) + (threadIdx.x >> 5);
    int lane = threadIdx.x & 31;
    if (e >= N_EDGES) return;
    int s = src[e];
    int d = dst[e];
    const float4 f = *(const float4*)(feature + (size_t)s * D + lane * 4);
    float* o = h + (size_t)d * D + lane * 4;
    unsafeAtomicAdd(o + 0, f.x);
    unsafeAtomicAdd(o + 1, f.y);
    unsafeAtomicAdd(o + 2, f.z);
    unsafeAtomicAdd(o + 3, f.w);
}

// ---------------------------------------------------------------------------
// Kernel 3: out = h @ W^T + b via V_WMMA_F32_16X16X4_F32, fully LDS-fed:
//   - W (64 KB) cooperatively staged to LDS once per block (B operand, =W^T)
//   - each wave's 16x128 A-tile staged with GLOBAL_LOAD_ASYNC_TO_LDS_B128
//     (16 coalesced 512B transfers, ASYNCcnt-tracked, s_wait_asynccnt)
//   A lane(l,hi):  h[m+l][k+2hi], h[m+l][k+2hi+1]   (ds_load_b64 from sH)
//   B lane(l,hi):  W[n+l][k+2hi], W[n+l][k+2hi+1]   (ds_load_b64 from sW)
//   D lane(l,hi) vgpr v -> out[m+v+8hi][n+l] ; bias folded into C init
// ---------------------------------------------------------------------------
__global__ void __launch_bounds__(256)
gcn_gemm(const float* __restrict__ h,
         const float* __restrict__ W,
         const float* __restrict__ bias,
         float* __restrict__ out) {
    __shared__ float sW[D * D];          // 64 KB, W row-major
    __shared__ float sH[8][16 * D];      // 64 KB, one 8 KB A-tile per wave
    {
        const float4* Wv  = (const float4*)W;
        float4*       sWv = (float4*)sW;
        for (int i = threadIdx.x; i < (D * D) / 4; i += blockDim.x)
            sWv[i] = Wv[i];
    }
    __syncthreads();

    const int wave = threadIdx.x >> 5;
    const int lane = threadIdx.x & 31;
    const int mt   = blockIdx.x * 8 + wave;
    if (mt >= MT_COUNT) return;          // wave-uniform: EXEC stays all-1s

    // ---- async-stage this wave's A-tile (contiguous 8 KB) into sH[wave] ----
    {
        const unsigned lds0 = (unsigned)(uintptr_t)&sH[wave][0]; // LDS byte offset
        const char*    g0   = (const char*)(h + (size_t)mt * 16 * D);
#pragma unroll
        for (int it = 0; it < 16; ++it) {
            unsigned    loff = lds0 + (unsigned)((it * 32 + lane) * 16);
            const char* ga   = g0 + (size_t)(it * 32 + lane) * 16;
            asm volatile("global_load_async_to_lds_b128 %0, %1, off"
                         :: "v"(loff), "v"(ga) : "memory");
        }
        asm volatile("s_wait_asynccnt 0x0" ::: "memory");
    }

    const int l  = lane & 15;
    const int hi = lane >> 4;
    const float* aRow = &sH[wave][l * D + 2 * hi];

    v8f acc[8];
#pragma unroll
    for (int nt = 0; nt < 8; ++nt) {
        const float bb = bias[nt * 16 + l]; // bias folded into C init
#pragma unroll
        for (int v = 0; v < 8; ++v) acc[nt][v] = bb;
    }

    for (int k = 0; k < D; k += 4) {
        const v2f a = *(const v2f*)(aRow + k);
#pragma unroll
        for (int nt = 0; nt < 8; ++nt) {
            const v2f bv = *(const v2f*)(&sW[(nt * 16 + l) * D + k + 2 * hi]);
            acc[nt] = __builtin_amdgcn_wmma_f32_16x16x4_f32(
                /*neg_a=*/false, a, /*neg_b=*/false, bv,
                /*c_mod=*/(short)0, acc[nt],
                /*reuse_a=*/false, /*reuse_b=*/false);
        }
    }

#pragma unroll
    for (int nt = 0; nt < 8; ++nt) {
#pragma unroll
        for (int v = 0; v < 8; ++v) {
            out[(size_t)(mt * 16 + v + 8 * hi) * D + nt * 16 + l] = acc[nt][v];
        }
    }
}

// ---------------------------------------------------------------------------
extern "C" void kernel_launch(void* const* d_in, const int* in_sizes, int n_in,
                              void* d_out, int out_size, void* d_ws, size_t ws_size,
                              hipStream_t stream) {
    const float* feature = (const float*)d_in[0];
    const int*   src     = (const int*)d_in[1];
    const int*   dst     = (const int*)d_in[2];
    const float* W       = (const float*)d_in[3];
    const float* bias    = (const float*)d_in[4];
    float*       out     = (float*)d_out;
    float*       h       = (float*)d_ws;   // N_NODES*D*4 = 25.6 MB scratch

    // 1) h = feature
    const int n4 = N_NODES * D / 4;
    gcn_init_h<<<(n4 + 255) / 256, 256, 0, stream>>>(
        (const float4*)feature, (float4*)h, n4);

    // 2) h[dst] += feature[src]
    const int epb = 256 / 32;              // 8 edges per 256-thread block
    gcn_scatter<<<(N_EDGES + epb - 1) / epb, 256, 0, stream>>>(
        feature, src, dst, h);

    // 3) out = h @ W^T + b
    gcn_gemm<<<(MT_COUNT + 7) / 8, 256, 0, stream>>>(h, W, bias, out);
}